// Pooler_53592601919893
// MI455X (gfx1250) — compile-verified
//
#include <hip/hip_runtime.h>
#include <cstdint>
#include <cstddef>

// Ragged mean pooling (Pooler, mode='mean'), B=8, T=4096, D=256, S=1024, W<=32.
// One wave32 per span. Window rows are contiguous in T, so each span streams a
// contiguous <=32KB block. Double-buffered CDNA5 async global->LDS staging
// (global_load_async_to_lds_b128 + s_wait_asynccnt), then ds_load_b128 +
// f32 accumulation. Memory-bound (~40MB HBM => ~1.7us at 23.3 TB/s); WMMA not
// applicable (batched mat-vec: the B operand differs per span).
//
// This revision moves all wave-uniform span metadata into the scalar pipe via
// readfirstlane: begins/ends become s_load, loop control becomes SALU, and the
// per-row global base is an SGPR pair advanced by a scalar add (feeding the
// "s" operand of the async op directly).

namespace {

constexpr int kB = 8;
constexpr int kT = 4096;
constexpr int kD = 256;     // floats per row (1 KB)
constexpr int kS = 1024;
constexpr int kMaxW = 32;
constexpr int kWavesPerBlock = 8;
constexpr uint32_t kRowBytes = kD * 4u;   // 1024

} // namespace

// Issue one full 1KB row as two async b128 loads per lane (offsets 0 and 512).
// INST_OFFSET is added on BOTH the LDS side and the global side (ISA 08 §4.4),
// so a single (lds_addr, voff) pair covers both halves. SBASE is a wave-
// uniform 64-bit row base held in an SGPR pair; VOFF is lane*16.
#define ISSUE_ROW_ASYNC(LDSA, VOFF, SBASE)                                  \
  asm volatile(                                                             \
      "global_load_async_to_lds_b128 %0, %1, %2 offset:0\n\t"               \
      "global_load_async_to_lds_b128 %0, %1, %2 offset:512"                 \
      :: "v"(LDSA), "v"(VOFF), "s"(SBASE) : "memory")

__global__ __launch_bounds__(256) void ragged_mean_kernel(
    const float* __restrict__ features,  // [B*T*D]
    const int*   __restrict__ begins,    // [B*S]
    const int*   __restrict__ ends,      // [B*S]
    float*       __restrict__ out)       // [B*S*D]
{
  // 8 waves * 2 row-buffers * 256 floats = 16 KB LDS per block.
  __shared__ __align__(16) float lbuf[kWavesPerBlock * 2 * kD];

  const int tid  = threadIdx.x;
  const int wave = tid >> 5;
  const int lane = tid & 31;

  // Span index is wave-uniform; force it scalar so the metadata fetches
  // become s_load_b32 and everything derived stays in SGPRs.
  const int gspan = __builtin_amdgcn_readfirstlane(
      (int)(blockIdx.x * kWavesPerBlock + wave));         // 0 .. B*S-1
  const int b = gspan >> 10;                              // / kS (kS == 1024)

  int bg = begins[gspan];
  int en = ends[gspan];
  bg = bg < 0 ? 0 : bg;
  en = en < 0 ? 0 : en;
  int cnt = en - bg;
  if (cnt < 0) cnt = 0;
  if (cnt > kMaxW) cnt = kMaxW;
  const int cnt_s = __builtin_amdgcn_readfirstlane(cnt);
  const int bg_s  = __builtin_amdgcn_readfirstlane(bg);

  // Wave-uniform 64-bit global base of the span's first row (SGPR pair).
  uint64_t rowbase = (uint64_t)(uintptr_t)features
                   + ((uint64_t)((uint32_t)(b * kT + bg_s)) * (uint64_t)kRowBytes);
  // Per-lane part of the address: this lane's 16B chunk within the row.
  const uint32_t voff = (uint32_t)lane * 16u;

  // LDS byte addresses of this wave's two row buffers (this lane's chunk).
  const int wbase_f = wave * (2 * kD);                    // float index of buf0
  uint32_t lds0;
  {
    // Strip the generic-pointer aperture: cast down to addrspace(3) so the
    // integer value is the raw LDS byte offset the async op expects.
    __attribute__((address_space(3))) const float* p3 =
        (__attribute__((address_space(3))) const float*)&lbuf[wbase_f];
    lds0 = (uint32_t)(uintptr_t)p3 + (uint32_t)lane * 16u;
  }
  const uint32_t lds1 = lds0 + kRowBytes;                 // buffer 1 (+1KB)

  float a0 = 0.f, a1 = 0.f, a2 = 0.f, a3 = 0.f;
  float a4 = 0.f, a5 = 0.f, a6 = 0.f, a7 = 0.f;

  if (cnt_s > 0) {                                        // scalar branch
    // Prologue: stage row 0 into buffer 0.
    ISSUE_ROW_ASYNC(lds0, voff, rowbase);
    uint64_t nextbase = rowbase + kRowBytes;              // scalar add

    for (int w = 0; w < cnt_s; ++w) {                     // SALU loop control
      if (w + 1 < cnt_s) {
        // Keep the pipe full: stage row w+1 into the other buffer, then wait
        // until only its 2 loads remain outstanding => row w has landed
        // (async completions are in-order per wave).
        const uint32_t lnxt = (w & 1) ? lds0 : lds1;
        ISSUE_ROW_ASYNC(lnxt, voff, nextbase);
        nextbase += kRowBytes;
        asm volatile("s_wait_asynccnt 2" ::: "memory");
      } else {
        asm volatile("s_wait_asynccnt 0" ::: "memory");
      }

      // Consume row w from its buffer: two b128 LDS reads per lane.
      const int fbase = wbase_f + (w & 1) * kD + lane * 4;
      const float4 va = *(const float4*)&lbuf[fbase];
      const float4 vb = *(const float4*)&lbuf[fbase + 128];
      a0 += va.x; a1 += va.y; a2 += va.z; a3 += va.w;
      a4 += vb.x; a5 += vb.y; a6 += vb.z; a7 += vb.w;
    }
  }

  const float inv = 1.0f / (float)(cnt_s > 0 ? cnt_s : 1);  // scalar-derived
  const size_t obase = (size_t)gspan * (size_t)kD + (size_t)lane * 4;
  float4 o0, o1;
  o0.x = a0 * inv; o0.y = a1 * inv; o0.z = a2 * inv; o0.w = a3 * inv;
  o1.x = a4 * inv; o1.y = a5 * inv; o1.z = a6 * inv; o1.w = a7 * inv;
  *(float4*)&out[obase]       = o0;
  *(float4*)&out[obase + 128] = o1;
}

extern "C" void kernel_launch(void* const* d_in, const int* in_sizes, int n_in,
                              void* d_out, int out_size, void* d_ws, size_t ws_size,
                              hipStream_t stream) {
  (void)in_sizes; (void)n_in; (void)out_size; (void)d_ws; (void)ws_size;
  const float* features = (const float*)d_in[0];
  const int*   begins   = (const int*)d_in[1];
  const int*   ends     = (const int*)d_in[2];
  float*       out      = (float*)d_out;

  const int total_spans = kB * kS;                        // 8192
  const int blocks = total_spans / kWavesPerBlock;        // 1024
  ragged_mean_kernel<<<blocks, 256, 0, stream>>>(features, begins, ends, out);
}